// MutualNearestNeighbor_17188459118800
// MI455X (gfx1250) — compile-verified
//
#include <hip/hip_runtime.h>
#include <stdint.h>

// ---------------------------------------------------------------------------
// Mutual nearest neighbor matching on MI455X (gfx1250), fused:
//   sim = d0^T (8192x256) * d1 (256x8192) in exact f32 via V_WMMA_F32_16X16X4_F32
//   A/B panels staged into LDS by the Tensor Data Mover (tensor_load_to_lds,
//   TENSORcnt-tracked, double-buffered), row/col argmax folded into the GEMM
//   epilogue with packed u64 atomic-max, tiny finalize kernel for the mutual
//   check + distance threshold.
// ---------------------------------------------------------------------------

typedef __attribute__((ext_vector_type(2))) float    v2f;
typedef __attribute__((ext_vector_type(8))) float    v8f;
typedef __attribute__((ext_vector_type(4))) uint32_t v4u;
typedef __attribute__((ext_vector_type(8))) uint32_t v8u;

#define D_DIM 256
#define N_DIM 8192
#define M_DIM 8192
#define BM 128            // tile rows (n)
#define BN 128            // tile cols (m)
#define KC 16             // k-chunk staged per TDM load
#define NCHUNK (D_DIM / KC)
#define LSTRIDE (BM + 8)  // 136 floats/row: 2*136 mod 64 == 16 -> halves hit disjoint banks

// order-preserving float->u32 encoding (monotone under unsigned compare)
__device__ __forceinline__ unsigned int enc_f32(float f) {
  unsigned int u = __float_as_uint(f);
  return (u & 0x80000000u) ? ~u : (u | 0x80000000u);
}
__device__ __forceinline__ float dec_f32(unsigned int e) {
  unsigned int u = (e & 0x80000000u) ? (e & 0x7FFFFFFFu) : ~e;
  return __uint_as_float(u);
}

// ---------------------------------------------------------------------------
// Tensor Data Mover: 2D tile load (global -> LDS), 2-group descriptor form.
// Tile: tile_d1 rows of tile_d0 f32, row stride `stride_elems` in memory.
// LDS layout: rows padded by 8 dwords every 128 dwords (pad_interval=6 -> 128
// dwords, pad_amount=7 -> 8 dwords), i.e. [row][136] floats.
// ---------------------------------------------------------------------------
__device__ __forceinline__ void tdm_load_tile_f32(const float* gptr, uint32_t lds_byte_addr,
                                                  uint32_t tile_d0, uint32_t tile_d1,
                                                  uint64_t stride_elems) {
  uint64_t ga = (uint64_t)(uintptr_t)gptr;
  v4u g0;
  g0[0] = 1u;                                              // count=1, is_restore=0, gather=0
  g0[1] = lds_byte_addr;                                   // lds_addr
  g0[2] = (uint32_t)(ga & 0xFFFFFFFFull);                  // global_addr[31:0]
  g0[3] = (uint32_t)((ga >> 32) & 0x01FFFFFFull)           // global_addr[56:32]
        | (2u << 30);                                      // type = 2 ("image")
  v8u g1;
  g1[0] = (2u << 16)                                       // data_size = 2 (4 bytes)
        | (1u << 20)                                       // pad_enable
        | (6u << 22)                                       // pad_interval: 128 dwords
        | (7u << 25);                                      // pad_amount:   8 dwords
  g1[1] = (tile_d0 & 0xFFFFu) << 16;                       // tensor_dim0[15:0] (tile always in-range)
  g1[2] = (tile_d0 >> 16) | ((tile_d1 & 0xFFFFu) << 16);   // tensor_dim0[31:16] | tensor_dim1[15:0]
  g1[3] = (tile_d1 >> 16) | (tile_d0 << 16);               // tensor_dim1[31:16] | tile_dim0
  g1[4] = tile_d1 & 0xFFFFu;                               // tile_dim1; tile_dim2 = 0
  g1[5] = (uint32_t)(stride_elems & 0xFFFFFFFFull);        // tensor_dim0_stride[31:0]
  g1[6] = (uint32_t)((stride_elems >> 32) & 0xFFFFull);    // tensor_dim0_stride[47:32]
  g1[7] = 0u;
  asm volatile("tensor_load_to_lds %0, %1" :: "s"(g0), "s"(g1) : "memory");
}

__global__ void mnn_init(unsigned long long* __restrict__ best, int n) {
  int i = blockIdx.x * blockDim.x + threadIdx.x;
  if (i < n) best[i] = 0ull;   // encoded sim >= -1.0 always beats 0
}

__global__ __launch_bounds__(256)
void mnn_sim_kernel(const float* __restrict__ d0,   // [D][N]
                    const float* __restrict__ d1,   // [D][M]
                    unsigned long long* __restrict__ rowbest,   // [N]
                    unsigned long long* __restrict__ colbest) { // [M]
  __shared__ float As[2][KC][LSTRIDE];   // As[buf][k][n], TDM pad reproduces LSTRIDE
  __shared__ float Bs[2][KC][LSTRIDE];   // Bs[buf][k][m]

  const int tid  = threadIdx.x;
  const int lane = tid & 31;
  const int w    = tid >> 5;          // wave 0..7
  const int wr   = w & 3;             // wave tile row (n): 4 x 32 rows
  const int wc   = w >> 2;            // wave tile col (m): 2 x 64 cols
  const int half = lane >> 4;
  const int lh   = lane & 15;

  const int n0 = blockIdx.y * BM;
  const int m0 = blockIdx.x * BN;

  const uint32_t ldsA0 = (uint32_t)(uintptr_t)&As[0][0][0];
  const uint32_t ldsA1 = (uint32_t)(uintptr_t)&As[1][0][0];
  const uint32_t ldsB0 = (uint32_t)(uintptr_t)&Bs[0][0][0];
  const uint32_t ldsB1 = (uint32_t)(uintptr_t)&Bs[1][0][0];

  v8f acc[2][4];
  #pragma unroll
  for (int tr = 0; tr < 2; ++tr)
    #pragma unroll
    for (int tc = 0; tc < 4; ++tc)
      #pragma unroll
      for (int i = 0; i < 8; ++i)
        acc[tr][tc][i] = 0.0f;

  // Prologue: TDM loads of chunk 0 into buffer 0 (wave 0 only; TDM ignores
  // EXEC, so gate with a wave-uniform branch -> issued exactly once).
  if (w == 0) {
    tdm_load_tile_f32(d0 + (size_t)0 * N_DIM + n0, ldsA0, BM, KC, N_DIM);
    tdm_load_tile_f32(d1 + (size_t)0 * M_DIM + m0, ldsB0, BN, KC, M_DIM);
  }

  for (int ic = 0; ic < NCHUNK; ++ic) {
    const int cur = ic & 1;
    if (w == 0) {
      if (ic + 1 < NCHUNK) {
        const size_t kn = (size_t)(ic + 1) * KC;
        tdm_load_tile_f32(d0 + kn * N_DIM + n0, cur ? ldsA0 : ldsA1, BM, KC, N_DIM);
        tdm_load_tile_f32(d1 + kn * M_DIM + m0, cur ? ldsB0 : ldsB1, BN, KC, M_DIM);
        // 4 outstanding TDM ops; retire the current chunk's two (in-order).
        __builtin_amdgcn_s_wait_tensorcnt(2);
      } else {
        __builtin_amdgcn_s_wait_tensorcnt(0);
      }
    }
    __syncthreads();   // chunk `ic` visible to all waves; next chunk streams in

    #pragma unroll
    for (int kk = 0; kk < KC; kk += 4) {
      // A 16x4 layout: lanes 0-15 hold K=kk,kk+1; lanes 16-31 hold K=kk+2,kk+3
      const int k0 = kk + 2 * half;
      v2f afrag[2], bfrag[4];
      #pragma unroll
      for (int tr = 0; tr < 2; ++tr) {
        int nl = wr * 32 + tr * 16 + lh;
        afrag[tr].x = As[cur][k0][nl];
        afrag[tr].y = As[cur][k0 + 1][nl];
      }
      #pragma unroll
      for (int tc = 0; tc < 4; ++tc) {
        int ml = wc * 64 + tc * 16 + lh;
        bfrag[tc].x = Bs[cur][k0][ml];
        bfrag[tc].y = Bs[cur][k0 + 1][ml];
      }
      #pragma unroll
      for (int tr = 0; tr < 2; ++tr)
        #pragma unroll
        for (int tc = 0; tc < 4; ++tc)
          acc[tr][tc] = __builtin_amdgcn_wmma_f32_16x16x4_f32(
              false, afrag[tr], false, bfrag[tc],
              (short)0, acc[tr][tc], false, false);
    }
    __syncthreads();   // done reading buf[cur] before TDM overwrites it (ic+2)
  }

  // ---- Epilogue: fused argmax reductions ----------------------------------
  // C/D layout: VGPR v, lane l -> row M = v + 8*half (+16*tr +32*wr), col N = lh (+16*tc +64*wc)

  // Row (n) argmax over m: pack (enc(sim) << 32) | ~m -> u64 max == (max sim, min m)
  #pragma unroll
  for (int tr = 0; tr < 2; ++tr) {
    #pragma unroll
    for (int vg = 0; vg < 8; ++vg) {
      unsigned long long best = 0ull;
      #pragma unroll
      for (int tc = 0; tc < 4; ++tc) {
        unsigned int m = (unsigned int)(m0 + wc * 64 + tc * 16 + lh);
        unsigned long long p =
            ((unsigned long long)enc_f32(acc[tr][tc][vg]) << 32) | (unsigned int)(~m);
        best = (p > best) ? p : best;
      }
      #pragma unroll
      for (int msk = 1; msk < 16; msk <<= 1) {
        unsigned long long o = __shfl_xor(best, msk, 32);
        best = (o > best) ? o : best;
      }
      if (lh == 0) {
        int row = n0 + wr * 32 + tr * 16 + half * 8 + vg;
        atomicMax(&rowbest[row], best);
      }
    }
  }

  // Col (m) argmax over n: lanes l and l+16 share a column -> combine via shfl_xor(16)
  #pragma unroll
  for (int tc = 0; tc < 4; ++tc) {
    unsigned long long best = 0ull;
    #pragma unroll
    for (int tr = 0; tr < 2; ++tr)
      #pragma unroll
      for (int vg = 0; vg < 8; ++vg) {
        unsigned int row = (unsigned int)(n0 + wr * 32 + tr * 16 + half * 8 + vg);
        unsigned long long p =
            ((unsigned long long)enc_f32(acc[tr][tc][vg]) << 32) | (unsigned int)(~row);
        best = (p > best) ? p : best;
      }
    {
      unsigned long long o = __shfl_xor(best, 16, 32);
      best = (o > best) ? o : best;
    }
    if (half == 0) {
      int col = m0 + wc * 64 + tc * 16 + lh;
      atomicMax(&colbest[col], best);
    }
  }
}

__global__ void mnn_finalize(const unsigned long long* __restrict__ rowbest,
                             const unsigned long long* __restrict__ colbest,
                             float* __restrict__ out) {
  int n = blockIdx.x * blockDim.x + threadIdx.x;
  if (n >= N_DIM) return;
  unsigned long long rp = rowbest[n];
  unsigned int m = ~(unsigned int)(rp & 0xFFFFFFFFull);
  float s = dec_f32((unsigned int)(rp >> 32));
  unsigned long long cp = colbest[m];
  unsigned int nb = ~(unsigned int)(cp & 0xFFFFFFFFull);
  float dist = 2.0f * (1.0f - s);
  bool ok = (nb == (unsigned int)n) && (dist <= 0.64f);  // 0.8^2
  out[n]         = ok ? (float)m : -1.0f;
  out[N_DIM + n] = ok ? (dist + 1.0f) * 0.5f : 0.0f;
}

extern "C" void kernel_launch(void* const* d_in, const int* in_sizes, int n_in,
                              void* d_out, int out_size, void* d_ws, size_t ws_size,
                              hipStream_t stream) {
  const float* d0 = (const float*)d_in[0];   // [256, 8192]
  const float* d1 = (const float*)d_in[1];   // [256, 8192]
  unsigned long long* rowbest = (unsigned long long*)d_ws;
  unsigned long long* colbest = rowbest + N_DIM;

  mnn_init<<<(N_DIM + M_DIM + 255) / 256, 256, 0, stream>>>(rowbest, N_DIM + M_DIM);

  dim3 grid(M_DIM / BN, N_DIM / BM);   // 64 x 64 tiles
  mnn_sim_kernel<<<grid, 256, 0, stream>>>(d0, d1, rowbest, colbest);

  mnn_finalize<<<(N_DIM + 255) / 256, 256, 0, stream>>>(rowbest, colbest, (float*)d_out);
}